// AvgPool2d_63539746177448
// MI455X (gfx1250) — compile-verified
//
#include <hip/hip_runtime.h>
#include <hip/hip_bf16.h>

// GEMM: out[64,4096] = enc_x[64,16384] @ toeplitz[4096,16384]^T  (all fp32)
// Memory-bound (~261 MB @ 23.3 TB/s => ~11us floor). Strategy:
//  - stream toeplitz with non-temporal loads, each row by the 4 M-group waves (keeps enc_x hot in L2)
//  - V_WMMA_F32_16X16X4_F32 with 2 N-tiles per wave to amortize the A load
//  - deterministic: each wave exclusively owns a 16x32 output strip, plain stores

typedef __attribute__((ext_vector_type(2))) float v2f;
typedef __attribute__((ext_vector_type(8))) float v8f;

#define MDIM 64
#define NDIM 4096
#define KDIM 16384

__global__ __launch_bounds__(256) void
avgpool_toeplitz_wmma_f32(const float* __restrict__ A,   // enc_x  [64][16384]
                          const float* __restrict__ Bt,  // toeplitz [4096][16384]
                          float* __restrict__ out) {     // [64][4096]
  const int lane = threadIdx.x & 31;
  const int wave = blockIdx.x * (blockDim.x >> 5) + (threadIdx.x >> 5); // 0..511
  const int mgroup = wave >> 7;   // 0..3   -> rows [mgroup*16, +16)
  const int ngroup = wave & 127;  // 0..127 -> cols [ngroup*32, +32)

  const int half = lane >> 4;     // 0: K={0,1}  1: K={2,3} (ISA 7.12.2 f32 A/B layout)
  const int l    = lane & 15;     // M row (A) / N col (B) within tile

  // Per-lane base pointers: one float2 per WMMA K-step of 4.
  const float* a_ptr  = A  + (size_t)(mgroup * 16 + l) * KDIM + (half << 1);
  const float* b0_ptr = Bt + (size_t)(ngroup * 32 + l) * KDIM + (half << 1);
  const float* b1_ptr = b0_ptr + (size_t)16 * KDIM;

  v8f acc0 = {};
  v8f acc1 = {};

#pragma unroll 8
  for (int k = 0; k < KDIM; k += 4) {
    v2f a  = *(const v2f*)(a_ptr + k);                              // reused: default RT
    v2f b0 = __builtin_nontemporal_load((const v2f*)(b0_ptr + k));  // streamed once: NT
    v2f b1 = __builtin_nontemporal_load((const v2f*)(b1_ptr + k));
    // D = A(16x4) x B(4x16) + C ; 8 args: neg_a, A, neg_b, B, c_mod, C, reuse_a, reuse_b
    acc0 = __builtin_amdgcn_wmma_f32_16x16x4_f32(false, a, false, b0,
                                                 (short)0, acc0, false, false);
    acc1 = __builtin_amdgcn_wmma_f32_16x16x4_f32(false, a, false, b1,
                                                 (short)0, acc1, false, false);
  }

  // C/D layout: VGPR r -> M = base + r (lanes 0-15) / base + 8 + r (lanes 16-31),
  //             N = lane & 15. Each wave owns its strip -> plain coalesced stores.
  const int row0 = mgroup * 16 + half * 8;
  const int col  = ngroup * 32 + l;
#pragma unroll
  for (int r = 0; r < 8; ++r) {
    out[(size_t)(row0 + r) * NDIM + col]      = acc0[r];
    out[(size_t)(row0 + r) * NDIM + col + 16] = acc1[r];
  }
}

extern "C" void kernel_launch(void* const* d_in, const int* in_sizes, int n_in,
                              void* d_out, int out_size, void* d_ws, size_t ws_size,
                              hipStream_t stream) {
  const float* enc_x    = (const float*)d_in[0];  // [64, 16384] fp32
  const float* toeplitz = (const float*)d_in[1];  // [4096, 16384] fp32
  float* out            = (float*)d_out;          // [64, 4096] fp32
  (void)in_sizes; (void)n_in; (void)out_size; (void)d_ws; (void)ws_size;

  // 512 waves total: 4 M-groups x 128 N-groups; 8 waves (256 thr) per block.
  dim3 grid(64), block(256);
  avgpool_toeplitz_wmma_f32<<<grid, block, 0, stream>>>(enc_x, toeplitz, out);
}